// SparseLayer_33380485825262
// MI455X (gfx1250) — compile-verified
//
#include <hip/hip_runtime.h>
#include <hip/hip_bf16.h>

#define N_LGN    17400
#define N_POST   50000
#define NNZ_CNT  800000
#define N_BASIS  5
#define N_RECEPT 10
#define SSEQ     128
#define ROW_PAD  8                         // rows padded to multiple of 8 (2 K-chunks)
#define NNZ_PAD  (NNZ_CNT + ROW_PAD * N_POST)

typedef __attribute__((ext_vector_type(2))) float v2f;
typedef __attribute__((ext_vector_type(8))) float v8f;

// ---------------- CSR construction ----------------

__global__ void init_rowptr(int* __restrict__ row_ptr) {
    int i = blockIdx.x * blockDim.x + threadIdx.x;
    if (i <= N_POST) row_ptr[i] = 0;
}

__global__ void hist_rows(const int* __restrict__ indices, int* __restrict__ row_ptr) {
    int e = blockIdx.x * blockDim.x + threadIdx.x;
    if (e < NNZ_CNT) atomicAdd(&row_ptr[indices[2 * e] + 1], 1);
}

// round each row's count up to a multiple of ROW_PAD (padding slots stay w=0)
__global__ void pad_counts(int* __restrict__ row_ptr) {
    int i = blockIdx.x * blockDim.x + threadIdx.x;
    if (i < N_POST) row_ptr[i + 1] = (row_ptr[i + 1] + (ROW_PAD - 1)) & ~(ROW_PAD - 1);
}

// single-workgroup inclusive scan over N_POST+1 entries (row_ptr[0]==0 on entry)
__global__ void scan_rowptr(int* __restrict__ data) {
    __shared__ int lds[1024];
    __shared__ int carry;
    if (threadIdx.x == 0) carry = 0;
    __syncthreads();
    const int n = N_POST + 1;
    for (int base = 0; base < n; base += 1024) {
        int i = base + threadIdx.x;
        int v = (i < n) ? data[i] : 0;
        lds[threadIdx.x] = v;
        __syncthreads();
        for (int off = 1; off < 1024; off <<= 1) {
            int t = (threadIdx.x >= off) ? lds[threadIdx.x - off] : 0;
            __syncthreads();
            lds[threadIdx.x] += t;
            __syncthreads();
        }
        int inc = lds[threadIdx.x] + carry;
        if (i < n) data[i] = inc;
        __syncthreads();
        if (threadIdx.x == 1023) carry = inc;
        __syncthreads();
    }
}

__global__ void copy_cursor(const int* __restrict__ row_ptr, int* __restrict__ cursor) {
    int i = blockIdx.x * blockDim.x + threadIdx.x;
    if (i < N_POST) cursor[i] = row_ptr[i];
}

__global__ void zero_edges(int2* __restrict__ edge) {
    int i = blockIdx.x * blockDim.x + threadIdx.x;
    if (i < NNZ_PAD) edge[i] = make_int2(0, 0);   // col=0, syn=0, w=+0.0f
}

// pack: meta = col | (syn << 20); w as raw bits — one b64 per edge
__global__ void scatter_edges(const int* __restrict__ indices,
                              const float* __restrict__ weights,
                              const int* __restrict__ syn_ids,
                              int* __restrict__ cursor,
                              int2* __restrict__ edge) {
    int e = blockIdx.x * blockDim.x + threadIdx.x;
    if (e < NNZ_CNT) {
        int r = indices[2 * e];
        int c = indices[2 * e + 1];
        int pos = atomicAdd(&cursor[r], 1);
        edge[pos] = make_int2(c | (syn_ids[e] << 20), __float_as_int(weights[e]));
    }
}

// xT[c*128 + s] = x[s*N_LGN + c] -> a gathered x-column is one contiguous 512B run
__global__ void transpose_x(const float* __restrict__ x, float* __restrict__ xT) {
    int idx = blockIdx.x * blockDim.x + threadIdx.x;
    if (idx < SSEQ * N_LGN) {
        int s = idx / N_LGN;
        int c = idx - s * N_LGN;
        xT[(size_t)c * SSEQ + s] = x[idx];
    }
}

// ---------------- WMMA SpMM ----------------
// One block per post neuron; 8 wave32s, each owns one 16-row s-block.
// out[s, post, r] = sum_e xT[col_e*128 + s] * (w_e * sw[syn_e, r])
// Rows padded to multiple of 8 -> fully branchless inner loop, 2 WMMAs/iter.
__global__ __launch_bounds__(256)
void spmm_wmma(const float* __restrict__ xT, const float* __restrict__ sw,
               const int* __restrict__ row_ptr, const int2* __restrict__ edge,
               float* __restrict__ out) {
    __shared__ float s_swp[N_RECEPT * 16];   // stride-16, zero-padded cols 5..15
    if (threadIdx.x < N_RECEPT * 16) {
        int syn = threadIdx.x >> 4;
        int col = threadIdx.x & 15;
        s_swp[threadIdx.x] = (col < N_BASIS) ? sw[syn * N_BASIS + col] : 0.f;
    }
    __syncthreads();

    const int post = blockIdx.x;
    const int wid  = threadIdx.x >> 5;   // s-block 0..7
    const int lane = threadIdx.x & 31;
    const int n16  = lane & 15;          // A: M index (s);  B/D: N index (basis)
    const int hi   = lane >> 4;          // K half: lanes 0-15 -> K0,K1; 16-31 -> K2,K3
    const int start = row_ptr[post];
    const int end   = row_ptr[post + 1];
    const int sRow  = wid * 16 + n16;

    v8f acc = {};
    for (int k = start; k < end; k += 2 * 4) {
        const int e0 = k + 2 * hi;
        // independent 8-byte edge records: issue all four loads up front
        int2 E0 = edge[e0];
        int2 E1 = edge[e0 + 1];
        int2 E2 = edge[e0 + 4];
        int2 E3 = edge[e0 + 5];

        int   c0 = E0.x & 0xFFFFF, sy0 = E0.x >> 20;
        int   c1 = E1.x & 0xFFFFF, sy1 = E1.x >> 20;
        int   c2 = E2.x & 0xFFFFF, sy2 = E2.x >> 20;
        int   c3 = E3.x & 0xFFFFF, sy3 = E3.x >> 20;

        float a0 = xT[(size_t)c0 * SSEQ + sRow];
        float a1 = xT[(size_t)c1 * SSEQ + sRow];
        float a2 = xT[(size_t)c2 * SSEQ + sRow];
        float a3 = xT[(size_t)c3 * SSEQ + sRow];

        float b0 = __int_as_float(E0.y) * s_swp[(sy0 << 4) + n16];
        float b1 = __int_as_float(E1.y) * s_swp[(sy1 << 4) + n16];
        float b2 = __int_as_float(E2.y) * s_swp[(sy2 << 4) + n16];
        float b3 = __int_as_float(E3.y) * s_swp[(sy3 << 4) + n16];

        v2f A0 = {a0, a1}, B0 = {b0, b1};
        v2f A1 = {a2, a3}, B1 = {b2, b3};
        acc = __builtin_amdgcn_wmma_f32_16x16x4_f32(false, A0, false, B0,
                                                    (short)0, acc, false, false);
        acc = __builtin_amdgcn_wmma_f32_16x16x4_f32(false, A1, false, B1,
                                                    (short)0, acc, false, false);
    }

    // D layout: lane holds N = lane%16; VGPR j holds M = j + 8*hi
    if (n16 < N_BASIS) {
        const size_t colBase = (size_t)post * N_BASIS + n16;
        #pragma unroll
        for (int j = 0; j < 8; ++j) {
            int s = wid * 16 + (j + hi * 8);
            out[(size_t)s * ((size_t)N_POST * N_BASIS) + colBase] = acc[j];
        }
    }
}

// ---------------- launch ----------------

extern "C" void kernel_launch(void* const* d_in, const int* in_sizes, int n_in,
                              void* d_out, int out_size, void* d_ws, size_t ws_size,
                              hipStream_t stream) {
    const float* inp     = (const float*)d_in[0];   // (1,128,17400) f32
    const int*   indices = (const int*)  d_in[1];   // (NNZ,2) i32
    const float* weights = (const float*)d_in[2];   // (NNZ,)  f32
    const float* sw      = (const float*)d_in[3];   // (10,5)  f32
    const int*   syn_ids = (const int*)  d_in[4];   // (NNZ,)  i32
    float* out = (float*)d_out;                     // (1,128,250000) f32

    // workspace layout: ~18.9 MB
    int*  row_ptr = (int*)d_ws;                              // N_POST+1
    int*  cursor  = row_ptr + (N_POST + 1);                  // N_POST
    int2* edge    = (int2*)(cursor + N_POST + 3);            // NNZ_PAD (8B-aligned)
    float* xT     = (float*)(edge + NNZ_PAD);                // 128*17400

    init_rowptr   <<<(N_POST + 256) / 256, 256, 0, stream>>>(row_ptr);
    hist_rows     <<<(NNZ_CNT + 255) / 256, 256, 0, stream>>>(indices, row_ptr);
    pad_counts    <<<(N_POST + 255) / 256, 256, 0, stream>>>(row_ptr);
    scan_rowptr   <<<1, 1024, 0, stream>>>(row_ptr);
    copy_cursor   <<<(N_POST + 255) / 256, 256, 0, stream>>>(row_ptr, cursor);
    zero_edges    <<<(NNZ_PAD + 255) / 256, 256, 0, stream>>>(edge);
    scatter_edges <<<(NNZ_CNT + 255) / 256, 256, 0, stream>>>(indices, weights, syn_ids,
                                                              cursor, edge);
    transpose_x   <<<(SSEQ * N_LGN + 255) / 256, 256, 0, stream>>>(inp, xT);
    spmm_wmma     <<<N_POST, 256, 0, stream>>>(xT, sw, row_ptr, edge, out);
}